// GraphSAGELayer_7043746365768
// MI455X (gfx1250) — compile-verified
//
#include <hip/hip_runtime.h>
#include <hip/hip_bf16.h>

// GraphSAGE layer for MI455X (gfx1250, wave32).
//   out = x @ W_self^T + b_self + (segsum(x[src], tgt)/max(deg,1)) @ W_neigh^T + b_neigh
//
// Phase 1: zero workspace (aggregate[N*D] + degree[N]).
// Phase 2: edge scatter-add (L2-resident fp32 atomics).
// Phase 3: fused mean-normalize + dual GEMM using V_WMMA_F32_16X16X4_F32.

#define NUM_NODES 50000
#define NUM_EDGES 800000
#define DIM 96

typedef __attribute__((ext_vector_type(2))) float v2f;
typedef __attribute__((ext_vector_type(8))) float v8f;

// ---------------------------------------------------------------- zero init
__global__ void sage_zero_kernel(float* __restrict__ p, int n) {
    int i = blockIdx.x * blockDim.x + threadIdx.x;
    if (i < n) p[i] = 0.0f;
}

// ---------------------------------------------------------------- scatter
// blockDim = (96, EPB): 96 feature lanes (3 waves of 32) per edge, EPB edges/block.
#define EPB 4
__global__ void __launch_bounds__(96 * EPB)
sage_scatter_kernel(const float* __restrict__ x,
                    const long long* __restrict__ edge_index,
                    float* __restrict__ agg,
                    float* __restrict__ deg) {
    const int e = blockIdx.x * EPB + threadIdx.y;
    const int d = threadIdx.x;
    const long long s = edge_index[e];              // source row
    const long long t = edge_index[NUM_EDGES + e];  // target row
    atomicAdd(&agg[(size_t)t * DIM + d], x[(size_t)s * DIM + d]);
    if (d == 0) atomicAdd(&deg[t], 1.0f);
}

// ---------------------------------------------------------------- fused GEMM
// Grid: 3125 blocks (16 rows each). Block: 192 threads = 6 waves; wave w owns
// output tile rows [m0, m0+16) x cols [16w, 16w+16).
// K accumulation: 24 WMMA steps over W_self with A = x, then 24 over W_neigh
// with A = agg * (1/max(deg,1)) (per-lane row scale, computed once).
//
// V_WMMA_F32_16X16X4_F32 operand layout (ISA 7.12.2):
//   A 16x4: lane L holds row M=L&15, float2 at K = 2*(L>>4).
//   B 4x16: lane L holds col N=L&15, float2 at K = 2*(L>>4)  (B[k][n]=W[n][k]
//           => contiguous float2 from row n of the row-major weight matrix).
//   C/D:    VGPR r, lanes 0-15 -> (M=r, N=lane); lanes 16-31 -> (M=8+r, N=lane-16).
__global__ void __launch_bounds__(192)
sage_gemm_kernel(const float* __restrict__ x,
                 const float* __restrict__ agg,
                 const float* __restrict__ deg,
                 const float* __restrict__ W_self,
                 const float* __restrict__ b_self,
                 const float* __restrict__ W_neigh,
                 const float* __restrict__ b_neigh,
                 float* __restrict__ out) {
    const int wave = threadIdx.x >> 5;      // 0..5  -> N tile
    const int lane = threadIdx.x & 31;
    const int l15  = lane & 15;
    const int kh   = (lane >> 4) << 1;      // 0 or 2: K sub-slice for this lane
    const int m0   = blockIdx.x << 4;       // 16 rows per block (50000 = 3125*16)
    const int n0   = wave << 4;
    const int m    = m0 + l15;              // A row this lane carries
    const int n    = n0 + l15;              // B column this lane carries

    const float rs = 1.0f / fmaxf(deg[m], 1.0f);   // mean normalizer for row m

    const float* __restrict__ arow  = x      + (size_t)m * DIM;
    const float* __restrict__ mrow  = agg    + (size_t)m * DIM;
    const float* __restrict__ bs_row = W_self  + (size_t)n * DIM;
    const float* __restrict__ bn_row = W_neigh + (size_t)n * DIM;

    v8f c = {};

    // ---- pass 1: x @ W_self^T  (24 fp32 WMMAs, K = 0..95 step 4)
    #pragma unroll
    for (int k = 0; k < DIM; k += 4) {
        v2f a = *(const v2f*)(arow + k + kh);
        v2f b = *(const v2f*)(bs_row + k + kh);
        c = __builtin_amdgcn_wmma_f32_16x16x4_f32(
                /*neg_a=*/false, a, /*neg_b=*/false, b,
                /*c_mod=*/(short)0, c, /*reuse_a=*/false, /*reuse_b=*/false);
    }

    // ---- pass 2: (agg * rs) @ W_neigh^T
    #pragma unroll
    for (int k = 0; k < DIM; k += 4) {
        v2f a = *(const v2f*)(mrow + k + kh);
        a.x *= rs;
        a.y *= rs;
        v2f b = *(const v2f*)(bn_row + k + kh);
        c = __builtin_amdgcn_wmma_f32_16x16x4_f32(
                false, a, false, b, (short)0, c, false, false);
    }

    // ---- bias + store (bias is per output column n, identical for all 8 rows)
    const float bias = b_self[n] + b_neigh[n];
    const int mbase = m0 + ((lane >> 4) << 3);     // +8 rows for upper half-wave
    #pragma unroll
    for (int r = 0; r < 8; ++r) {
        out[(size_t)(mbase + r) * DIM + n] = c[r] + bias;
    }
}

// ---------------------------------------------------------------- launcher
extern "C" void kernel_launch(void* const* d_in, const int* in_sizes, int n_in,
                              void* d_out, int out_size, void* d_ws, size_t ws_size,
                              hipStream_t stream) {
    const float*     x       = (const float*)d_in[0];
    const long long* eidx    = (const long long*)d_in[1];
    const float*     W_self  = (const float*)d_in[2];
    const float*     b_self  = (const float*)d_in[3];
    const float*     W_neigh = (const float*)d_in[4];
    const float*     b_neigh = (const float*)d_in[5];
    float*           out     = (float*)d_out;

    float* agg = (float*)d_ws;                       // [NUM_NODES * DIM]
    float* deg = agg + (size_t)NUM_NODES * DIM;      // [NUM_NODES]

    // 1) zero aggregate + degree
    const int nz = NUM_NODES * DIM + NUM_NODES;
    sage_zero_kernel<<<(nz + 255) / 256, 256, 0, stream>>>(agg, nz);

    // 2) edge scatter-add (800000 edges, 4 per block, 96 lanes per edge)
    sage_scatter_kernel<<<NUM_EDGES / EPB, dim3(DIM, EPB), 0, stream>>>(
        x, eidx, agg, deg);

    // 3) fused mean + dual GEMM + bias via fp32 WMMA
    sage_gemm_kernel<<<NUM_NODES / 16, 192, 0, stream>>>(
        x, agg, deg, W_self, b_self, W_neigh, b_neigh, out);
}